// MetaGraphGNN_59931973649027
// MI455X (gfx1250) — compile-verified
//
#include <hip/hip_runtime.h>
#include <hip/hip_bf16.h>

#define DIM 64
#define HEADS 4
#define HD 256          // HEADS*DIM
#define NEG_SLOPE 0.2f

typedef __attribute__((ext_vector_type(2))) float v2f;
typedef __attribute__((ext_vector_type(8))) float v8f;

// Monotone float<->uint mapping so unsigned atomicMax implements float max.
__device__ __forceinline__ unsigned f2ord(float f) {
    unsigned u = __float_as_uint(f);
    return (u & 0x80000000u) ? ~u : (u | 0x80000000u);
}
__device__ __forceinline__ float ord2f(unsigned u) {
    u = (u & 0x80000000u) ? (u & 0x7FFFFFFFu) : ~u;
    return __uint_as_float(u);
}

// ---------------- K0: init output with bias, reset softmax state ------------
__global__ void gat_init(float* __restrict__ out, const float* __restrict__ bias,
                         unsigned* __restrict__ segmax, float* __restrict__ segsum,
                         int N) {
    int i = blockIdx.x * blockDim.x + threadIdx.x;
    if (i < N * DIM) out[i] = bias[i & (DIM - 1)];
    if (i < N * HEADS) { segmax[i] = 0u; segsum[i] = 0.0f; }
}

// ---------------- K1: XW = X @ W via V_WMMA_F32_16X16X4_F32 -----------------
// grid.x = row tiles of 16, grid.y = 2 (two 128-col halves), block = 256 (8 waves)
// Each wave owns a 16x16 output tile; K=64 -> 16 WMMA steps of K=4.
__global__ void gat_gemm_wmma(const float* __restrict__ X, const float* __restrict__ W,
                              float* __restrict__ XW, int N) {
    const int row0 = blockIdx.x * 16;
    const int wave = threadIdx.x >> 5;
    const int lane = threadIdx.x & 31;
    const int col0 = blockIdx.y * 128 + wave * 16;
    const int lm = lane & 15;     // M (A) / N (B,C) position
    const int lh = lane >> 4;     // half-wave selector

    int arow = row0 + lm;
    arow = arow < N ? arow : N - 1;           // clamp (no EXEC divergence)
    const float* __restrict__ xr = X + (size_t)arow * DIM;

    v8f acc = {};
    #pragma unroll
    for (int k0 = 0; k0 < DIM; k0 += 4) {
        // A 16x4 fp32: lane l, vgpr v -> (M = l%16, K = k0 + 2*(l/16) + v)
        v2f a;
        a.x = xr[k0 + 2 * lh + 0];
        a.y = xr[k0 + 2 * lh + 1];
        // B 4x16 fp32: lane l, vgpr v -> (K = k0 + 2*(l/16) + v, N = l%16)
        v2f b;
        b.x = W[(size_t)(k0 + 2 * lh + 0) * HD + col0 + lm];
        b.y = W[(size_t)(k0 + 2 * lh + 1) * HD + col0 + lm];
        acc = __builtin_amdgcn_wmma_f32_16x16x4_f32(
            /*neg_a=*/false, a, /*neg_b=*/false, b,
            /*c_mod=*/(short)0, acc, /*reuse_a=*/false, /*reuse_b=*/false);
    }
    // C/D 16x16 f32: vgpr v, lane l -> row = v + 8*(l/16), col = l%16
    #pragma unroll
    for (int v = 0; v < 8; ++v) {
        int r = row0 + v + 8 * lh;
        if (r < N) XW[(size_t)r * HD + col0 + lm] = acc[v];
    }
}

// ---------------- K2: per-node attention scores -----------------------------
__global__ void gat_scores(const float* __restrict__ XW,
                           const float* __restrict__ att_src,
                           const float* __restrict__ att_dst,
                           float* __restrict__ a_src, float* __restrict__ a_dst,
                           int N) {
    int i = blockIdx.x * blockDim.x + threadIdx.x;   // over N*HEADS
    if (i >= N * HEADS) return;
    int n = i >> 2, h = i & (HEADS - 1);
    const float* __restrict__ xp = XW + (size_t)n * HD + h * DIM;
    const float* __restrict__ as = att_src + h * DIM;
    const float* __restrict__ ad = att_dst + h * DIM;
    float s1 = 0.f, s2 = 0.f;
    #pragma unroll
    for (int c = 0; c < DIM; ++c) {
        float x = xp[c];
        s1 += x * as[c];
        s2 += x * ad[c];
    }
    a_src[i] = s1;
    a_dst[i] = s2;
}

// ---------------- K3: edge logits + leaky-relu + segment max ---------------
__global__ void gat_edge_max(const int* __restrict__ edge,
                             const float* __restrict__ a_src,
                             const float* __restrict__ a_dst,
                             float* __restrict__ logits,
                             unsigned* __restrict__ segmax,
                             int E, int N) {
    int e = blockIdx.x * blockDim.x + threadIdx.x;
    if (e >= E + N) return;
    int s, d;
    if (e < E) { s = edge[e]; d = edge[E + e]; } else { s = d = e - E; }
    s = min(max(s, 0), N - 1);
    d = min(max(d, 0), N - 1);
    #pragma unroll
    for (int h = 0; h < HEADS; ++h) {
        float l = a_src[s * HEADS + h] + a_dst[d * HEADS + h];
        l = l > 0.f ? l : NEG_SLOPE * l;
        logits[(size_t)e * HEADS + h] = l;
        atomicMax(&segmax[d * HEADS + h], f2ord(l));
    }
}

// ---------------- K4: exp(logit - max), segment sum (overwrites logits) ----
__global__ void gat_edge_expsum(const int* __restrict__ edge,
                                const unsigned* __restrict__ segmax,
                                float* __restrict__ logits,
                                float* __restrict__ segsum,
                                int E, int N) {
    int e = blockIdx.x * blockDim.x + threadIdx.x;
    if (e >= E + N) return;
    int d = (e < E) ? edge[E + e] : (e - E);
    d = min(max(d, 0), N - 1);
    #pragma unroll
    for (int h = 0; h < HEADS; ++h) {
        float l = logits[(size_t)e * HEADS + h];
        float m = ord2f(segmax[d * HEADS + h]);
        float ex = __expf(l - m);
        logits[(size_t)e * HEADS + h] = ex;     // becomes ex
        atomicAdd(&segsum[d * HEADS + h], ex);
    }
}

// ---------------- K5: normalize + aggregate + head-mean into out ------------
// 64 threads per edge (one per channel), 4 edges per 256-thread block.
__global__ void gat_edge_aggregate(const int* __restrict__ edge,
                                   const float* __restrict__ XW,
                                   const float* __restrict__ ex,     // logits buf
                                   const float* __restrict__ segsum,
                                   float* __restrict__ out,
                                   int E, int N) {
    int t = threadIdx.x;
    int e = blockIdx.x * 4 + (t >> 6);
    int c = t & (DIM - 1);
    if (e >= E + N) return;
    int s, d;
    if (e < E) { s = edge[e]; d = edge[E + e]; } else { s = d = e - E; }
    s = min(max(s, 0), N - 1);
    d = min(max(d, 0), N - 1);
    const float* __restrict__ xs = XW + (size_t)s * HD;
    float contrib = 0.f;
    #pragma unroll
    for (int h = 0; h < HEADS; ++h) {
        float alpha = ex[(size_t)e * HEADS + h] /
                      (segsum[d * HEADS + h] + 1e-16f);
        contrib += alpha * xs[h * DIM + c];
    }
    atomicAdd(&out[(size_t)d * DIM + c], 0.25f * contrib);
}

// ---------------------------------------------------------------------------
extern "C" void kernel_launch(void* const* d_in, const int* in_sizes, int n_in,
                              void* d_out, int out_size, void* d_ws, size_t ws_size,
                              hipStream_t stream) {
    (void)n_in; (void)out_size; (void)ws_size;
    const float* X       = (const float*)d_in[0];
    const int*   edge    = (const int*)d_in[1];
    const float* W       = (const float*)d_in[2];
    const float* att_src = (const float*)d_in[3];
    const float* att_dst = (const float*)d_in[4];
    const float* bias    = (const float*)d_in[5];

    const int N = in_sizes[0] / DIM;
    const int E = in_sizes[1] / 2;
    const int ET = E + N;          // with self-loops

    float* out = (float*)d_out;

    // Workspace layout (all 4-byte elements):
    float*    xw     = (float*)d_ws;                 // N * 256
    float*    a_src  = xw + (size_t)N * HD;          // N * 4
    float*    a_dst  = a_src + (size_t)N * HEADS;    // N * 4
    unsigned* segmax = (unsigned*)(a_dst + (size_t)N * HEADS); // N * 4
    float*    segsum = (float*)(segmax + (size_t)N * HEADS);   // N * 4
    float*    logits = segsum + (size_t)N * HEADS;   // (E+N) * 4

    const int B = 256;

    gat_init<<<dim3((N * DIM + B - 1) / B), dim3(B), 0, stream>>>(
        out, bias, segmax, segsum, N);

    gat_gemm_wmma<<<dim3((N + 15) / 16, 2), dim3(B), 0, stream>>>(X, W, xw, N);

    gat_scores<<<dim3((N * HEADS + B - 1) / B), dim3(B), 0, stream>>>(
        xw, att_src, att_dst, a_src, a_dst, N);

    gat_edge_max<<<dim3((ET + B - 1) / B), dim3(B), 0, stream>>>(
        edge, a_src, a_dst, logits, segmax, E, N);

    gat_edge_expsum<<<dim3((ET + B - 1) / B), dim3(B), 0, stream>>>(
        edge, segmax, logits, segsum, E, N);

    gat_edge_aggregate<<<dim3((ET + 3) / 4), dim3(B), 0, stream>>>(
        edge, xw, logits, segsum, out, E, N);
}